// DSBatchNorm_4690104287826
// MI455X (gfx1250) — compile-verified
//
#include <hip/hip_runtime.h>

typedef __attribute__((ext_vector_type(2))) float v2f;
typedef __attribute__((ext_vector_type(4))) float v4f;
typedef __attribute__((ext_vector_type(8))) float v8f;

#define NFEAT 256
#define NDOM  8

// Workspace layout (floats):
//  [0,    2048)  s1[8][256]
//  [2048, 4096)  s2[8][256]
//  [4096, 4112)  counts (8 used, padded to 16 to keep scale float4-aligned)
//  [4112, 6160)  scale[8][256]
//  [6160, 8208)  shift[8][256]
#define WS_S1    0
#define WS_S2    2048
#define WS_CNT   4096
#define WS_SCALE 4112
#define WS_SHIFT 6160
#define WS_ZERO_FLOATS 4112

__global__ __launch_bounds__(256) void ds_bn_zero(float* __restrict__ ws, int nfloats) {
    int i = blockIdx.x * 256 + threadIdx.x;
    if (i < nfloats) ws[i] = 0.0f;
}

// Pass 1: segment sums s1 = onehot(y)^T * X and s2 = onehot(y)^T * (X*X)
// expressed as V_WMMA_F32_16X16X4_F32 (exact fp32, one-hot rows are exact).
// Each wave owns two 16-wide feature tiles; 8 waves per block cover F=256.
__global__ __launch_bounds__(256) void ds_bn_pass1(const float* __restrict__ x,
                                                   const int* __restrict__ y,
                                                   float* __restrict__ ws,
                                                   int rows_per_wg, int n) {
    float* s1  = ws + WS_S1;
    float* s2  = ws + WS_S2;
    float* cnt = ws + WS_CNT;

    const int tid  = threadIdx.x;
    const int wave = tid >> 5;
    const int lane = tid & 31;

    int row0    = blockIdx.x * rows_per_wg;
    int row_end = row0 + rows_per_wg;
    if (row_end > n) row_end = n;
    if (row0 >= n) return;

    // Per-block domain counts via LDS bins.
    __shared__ float lcnt[NDOM];
    if (tid < NDOM) lcnt[tid] = 0.0f;
    __syncthreads();
    for (int r = row0 + tid; r < row_end; r += 256)
        atomicAdd(&lcnt[y[r]], 1.0f);

    // WMMA operand geometry (wave32, 32-bit A 16x4 / B 4x16 layouts):
    //  A: lanes 0-15 hold K={0,1}, lanes 16-31 hold K={2,3}; M = lane&15
    //  B: same K split per lane half;                        N = lane&15
    const int  col0   = wave * 32;        // this wave's two feature tiles
    const int  m      = lane & 15;
    const bool hi     = (lane >> 4) != 0; // upper lane half -> K pair {2,3}
    const int  khalf  = hi ? 2 : 0;
    const int  ncol   = lane & 15;

    v8f c1a = {}, c2a = {}, c1b = {}, c2b = {};

    int rmain = row0 + ((row_end - row0) & ~3);
    for (int r = row0; r < rmain; r += 4) {
        // Uniform address -> scalar load (s_load_b128) of the 4 labels for
        // this K-group; lane-half selection stays in VALU cndmasks.
        const int4 yv = *reinterpret_cast<const int4*>(y + r);
        int ya = hi ? yv.z : yv.x;
        int yb = hi ? yv.w : yv.y;
        v2f a;
        a.x = (ya == m) ? 1.0f : 0.0f;
        a.y = (yb == m) ? 1.0f : 0.0f;

        const float* pa = x + (size_t)(r + khalf) * NFEAT + col0 + ncol;
        const float* pb = x + (size_t)(r + khalf + 1) * NFEAT + col0 + ncol;
        v2f b0, b1;
        b0.x = pa[0];   b0.y = pb[0];
        b1.x = pa[16];  b1.y = pb[16];
        v2f b0s = b0 * b0;
        v2f b1s = b1 * b1;

        c1a = __builtin_amdgcn_wmma_f32_16x16x4_f32(false, a, false, b0,  (short)0, c1a, false, false);
        c2a = __builtin_amdgcn_wmma_f32_16x16x4_f32(false, a, false, b0s, (short)0, c2a, false, false);
        c1b = __builtin_amdgcn_wmma_f32_16x16x4_f32(false, a, false, b1,  (short)0, c1b, false, false);
        c2b = __builtin_amdgcn_wmma_f32_16x16x4_f32(false, a, false, b1s, (short)0, c2b, false, false);
    }

    // Scalar tail (<=3 rows), whole block cooperates.
    for (int r = rmain; r < row_end; ++r) {
        int d = y[r];
        for (int f = tid; f < NFEAT; f += 256) {
            float v = x[(size_t)r * NFEAT + f];
            atomicAdd(&s1[d * NFEAT + f], v);
            atomicAdd(&s2[d * NFEAT + f], v * v);
        }
    }

    // Flush accumulators. C/D layout: VGPR mr, lanes 0-15 -> M=mr (domains 0..7),
    // lanes 16-31 -> M=mr+8 which is always zero here (y < 8), so skip them.
    if (lane < 16) {
#pragma unroll
        for (int mr = 0; mr < 8; ++mr) {
            atomicAdd(&s1[mr * NFEAT + col0 + lane],      c1a[mr]);
            atomicAdd(&s2[mr * NFEAT + col0 + lane],      c2a[mr]);
            atomicAdd(&s1[mr * NFEAT + col0 + 16 + lane], c1b[mr]);
            atomicAdd(&s2[mr * NFEAT + col0 + 16 + lane], c2b[mr]);
        }
    }

    __syncthreads();
    if (tid < NDOM) atomicAdd(&cnt[tid], lcnt[tid]);
}

// Pass 2: fold stats into per-(domain,feature) scale/shift.
__global__ __launch_bounds__(NFEAT) void ds_bn_pass2(const float* __restrict__ gamma,
                                                     const float* __restrict__ beta,
                                                     float* __restrict__ ws) {
    int d = blockIdx.x;   // 0..7
    int f = threadIdx.x;  // 0..255
    const float* s1  = ws + WS_S1;
    const float* s2  = ws + WS_S2;
    const float* cnt = ws + WS_CNT;
    float* scale = ws + WS_SCALE;
    float* shift = ws + WS_SHIFT;

    float c    = cnt[d];
    float cm   = fmaxf(c, 1.0f);
    float mean = s1[d * NFEAT + f] / cm;
    float var  = s2[d * NFEAT + f] / cm - mean * mean;
    float inv  = rsqrtf(var + 1e-5f);

    float sc = 1.0f, sh = 0.0f;   // identity when domain has <=1 sample
    if (c > 1.0f) {
        sc = gamma[d * NFEAT + f] * inv;
        sh = beta[d * NFEAT + f] - mean * sc;
    }
    scale[d * NFEAT + f] = sc;
    shift[d * NFEAT + f] = sh;
}

// Pass 3: out = x * scale[y] + shift[y], float4-vectorized, fully coalesced.
__global__ __launch_bounds__(256) void ds_bn_pass3(const float* __restrict__ x,
                                                   const int* __restrict__ y,
                                                   const float* __restrict__ ws,
                                                   float* __restrict__ out, int n) {
    const v4f* xv  = (const v4f*)x;
    v4f*       ov  = (v4f*)out;
    const v4f* sc4 = (const v4f*)(ws + WS_SCALE);
    const v4f* sh4 = (const v4f*)(ws + WS_SHIFT);

    size_t idx   = (size_t)blockIdx.x * 256 + threadIdx.x;
    size_t total = (size_t)n * (NFEAT / 4);
    if (idx >= total) return;

    size_t row = idx >> 6;          // 64 float4 per row
    int    fq  = (int)(idx & 63);
    int    d   = y[row];

    v4f xr = xv[idx];
    v4f sc = sc4[d * 64 + fq];
    v4f sh = sh4[d * 64 + fq];
    ov[idx] = xr * sc + sh;
}

extern "C" void kernel_launch(void* const* d_in, const int* in_sizes, int n_in,
                              void* d_out, int out_size, void* d_ws, size_t ws_size,
                              hipStream_t stream) {
    const float* x     = (const float*)d_in[0];
    const int*   y     = (const int*)d_in[1];
    const float* gamma = (const float*)d_in[2];
    const float* beta  = (const float*)d_in[3];
    float* out = (float*)d_out;
    float* ws  = (float*)d_ws;

    int n = in_sizes[1];  // y has N elements

    // Reset accumulators (ws is not re-poisoned between replays; this keeps
    // every call deterministic).
    ds_bn_zero<<<(WS_ZERO_FLOATS + 255) / 256, 256, 0, stream>>>(ws, WS_ZERO_FLOATS);

    int nwg = 512;
    int rows_per_wg = (n + nwg - 1) / nwg;
    rows_per_wg = (rows_per_wg + 3) & ~3;          // keep the WMMA path dominant
    nwg = (n + rows_per_wg - 1) / rows_per_wg;
    ds_bn_pass1<<<nwg, 256, 0, stream>>>(x, y, ws, rows_per_wg, n);

    ds_bn_pass2<<<NDOM, NFEAT, 0, stream>>>(gamma, beta, ws);

    size_t total4 = (size_t)n * (NFEAT / 4);
    int blocks3 = (int)((total4 + 255) / 256);
    ds_bn_pass3<<<blocks3, 256, 0, stream>>>(x, y, ws, out, n);
}